// Mamba3SISO_50646254354901
// MI455X (gfx1250) — compile-verified
//
#include <hip/hip_runtime.h>
#include <math.h>

// Problem constants (from reference)
#define B_   2
#define L_   2048
#define DM   1024     // D_MODEL
#define DIP  2432     // D_IN_PROJ
#define DI   2048     // D_INNER
#define NH   32       // NHEADS
#define HD   64       // HEADDIM
#define DS   128      // D_STATE
#define NR   32       // NROPE
#define CH   64       // CHUNK
#define NC   32       // L_/CH

typedef __attribute__((ext_vector_type(2))) float v2f;
typedef __attribute__((ext_vector_type(8))) float v8f;

static __device__ __forceinline__ v8f wmma4(v2f a, v2f b, v8f c) {
  // D = A(16x4 f32) * B(4x16 f32) + C(16x16 f32)
  return __builtin_amdgcn_wmma_f32_16x16x4_f32(false, a, false, b, (short)0, c,
                                               false, false);
}

static __device__ __forceinline__ v8f v8zero() {
  v8f z = {0.f, 0.f, 0.f, 0.f, 0.f, 0.f, 0.f, 0.f};
  return z;
}

static __device__ __forceinline__ float softplus_f(float x) {
  return (x > 20.f) ? x : log1pf(expf(x));
}

// -------------------------------------------------------------------------
// Generic WMMA fp32 GEMM: C[M,N] = A[M,K] * Wt[N,K]^T   (both row-major)
// 128x64 tile per 128-thread (4-wave) workgroup; each wave owns TWO 16-row
// strips so every B fragment is reused twice (0.75 LDS loads per WMMA).
// All LDS tiles in natural k-contiguous layout -> fragments are single
// aligned 8-byte LDS reads.
// -------------------------------------------------------------------------
__global__ __launch_bounds__(128) void gemm_nt_kernel(
    const float* __restrict__ A, const float* __restrict__ Wt,
    float* __restrict__ C, int M, int N, int K) {
  __shared__ float As[128][68];  // As[m][k]
  __shared__ float Bs[64][68];   // Bs[n][k]  (row of Wt, k-contiguous)
  const int tid  = threadIdx.x;
  const int wave = tid >> 5, lane = tid & 31;
  const int lm   = lane & 15;
  const int kg   = (lane >> 4) << 1;   // 0 or 2
  const int hi8  = (lane >> 4) << 3;   // 0 or 8
  const int m0 = blockIdx.y * 128;
  const int n0 = blockIdx.x * 64;

  v8f acc[2][4];
#pragma unroll
  for (int s = 0; s < 2; ++s)
#pragma unroll
    for (int t = 0; t < 4; ++t) acc[s][t] = v8zero();

  for (int kt = 0; kt < K; kt += 64) {
    // vectorized global -> LDS staging (float4)
#pragma unroll
    for (int i = 0; i < 16; ++i) {   // A tile: 128 rows x 16 float4
      int idx = tid + i * 128;
      int r = idx >> 4, c4 = idx & 15;
      *(float4*)(&As[r][c4 * 4]) =
          *(const float4*)(A + (size_t)(m0 + r) * K + kt + c4 * 4);
    }
#pragma unroll
    for (int i = 0; i < 8; ++i) {    // B tile: 64 rows x 16 float4
      int idx = tid + i * 128;
      int r = idx >> 4, c4 = idx & 15;
      *(float4*)(&Bs[r][c4 * 4]) =
          *(const float4*)(Wt + (size_t)(n0 + r) * K + kt + c4 * 4);
    }
    __syncthreads();
#pragma unroll 4
    for (int kk = 0; kk < 64; kk += 4) {
      v2f a0 = *(const v2f*)(&As[wave * 16 + lm][kk + kg]);
      v2f a1 = *(const v2f*)(&As[64 + wave * 16 + lm][kk + kg]);
#pragma unroll
      for (int t = 0; t < 4; ++t) {
        v2f bf = *(const v2f*)(&Bs[t * 16 + lm][kk + kg]);
        acc[0][t] = wmma4(a0, bf, acc[0][t]);
        acc[1][t] = wmma4(a1, bf, acc[1][t]);
      }
    }
    __syncthreads();
  }
#pragma unroll
  for (int s = 0; s < 2; ++s)
#pragma unroll
    for (int t = 0; t < 4; ++t)
#pragma unroll
      for (int i = 0; i < 8; ++i) {
        int m = m0 + s * 64 + wave * 16 + i + hi8;
        int n = n0 + t * 16 + lm;
        C[(size_t)m * N + n] = acc[s][t][i];
      }
}

// -------------------------------------------------------------------------
// Per-token prep: rmsnorm(Braw/Craw), softplus chain -> la, DT, w1, w2
// -------------------------------------------------------------------------
__global__ __launch_bounds__(128) void prep_token_kernel(
    const float* __restrict__ proj, const float* __restrict__ dt_bias,
    const float* __restrict__ Bnw, const float* __restrict__ Cnw,
    float* __restrict__ Brms, float* __restrict__ Crms,
    float* __restrict__ laB, float* __restrict__ DTb,
    float* __restrict__ w1b, float* __restrict__ w2b) {
  __shared__ float sB[128], sC[128];
  const int tok = blockIdx.x;  // b*L + l
  const int tid = threadIdx.x;
  const float* row = proj + (size_t)tok * DIP;
  float br = row[DI + tid];
  float cr = row[DI + DS + tid];
  sB[tid] = br * br;
  sC[tid] = cr * cr;
  __syncthreads();
  for (int s = 64; s > 0; s >>= 1) {
    if (tid < s) { sB[tid] += sB[tid + s]; sC[tid] += sC[tid + s]; }
    __syncthreads();
  }
  float rb = rsqrtf(sB[0] * (1.0f / DS) + 1e-5f);
  float rc = rsqrtf(sC[0] * (1.0f / DS) + 1e-5f);
  Brms[(size_t)tok * DS + tid] = br * rb * Bnw[tid];
  Crms[(size_t)tok * DS + tid] = cr * rc * Cnw[tid];
  if (tid < NH) {
    int h = tid;
    float ddt  = row[DI + 2 * DS + h];
    float dA   = row[DI + 2 * DS + NH + h];
    float trap = row[DI + 2 * DS + 2 * NH + h];
    float Aa = -softplus_f(dA);
    Aa = fminf(Aa, -1e-4f);
    float DT = softplus_f(ddt + dt_bias[h]);
    float la = Aa * DT;
    float lam = 1.0f / (1.0f + expf(-trap));
    size_t o = (size_t)tok * NH + h;
    laB[o] = la;
    DTb[o] = DT;
    w1b[o] = DT * lam;
    w2b[o] = DT * (1.0f - lam) * expf(la);
  }
}

// -------------------------------------------------------------------------
// Theta = cumsum_l(angles * DT); store cos/sin.  One thread per (b,h,r).
// -------------------------------------------------------------------------
__global__ void theta_kernel(const float* __restrict__ proj,
                             const float* __restrict__ DTb,
                             float* __restrict__ cosT, float* __restrict__ sinT) {
  int id = blockIdx.x * blockDim.x + threadIdx.x;
  if (id >= B_ * NH * NR) return;
  int r = id % NR;
  int h = (id / NR) % NH;
  int b = id / (NR * NH);
  float acc = 0.f;
  for (int l = 0; l < L_; ++l) {
    size_t tok = (size_t)b * L_ + l;
    float ang = proj[tok * DIP + DI + 2 * DS + 3 * NH + r];
    float dt = DTb[tok * NH + h];
    acc += ang * dt;
    size_t o = (tok * NH + h) * NR + r;
    cosT[o] = cosf(acc);
    sinT[o] = sinf(acc);
  }
}

// -------------------------------------------------------------------------
// Per-chunk cumsum of la -> Acum; cd = exp(Acum[last]).  Thread per (b,z,h).
// -------------------------------------------------------------------------
__global__ void acum_kernel(const float* __restrict__ laB,
                            float* __restrict__ Acum, float* __restrict__ cd) {
  int id = blockIdx.x * blockDim.x + threadIdx.x;
  if (id >= B_ * NC * NH) return;
  int h = id % NH;
  int z = (id / NH) % NC;
  int b = id / (NH * NC);
  float acc = 0.f;
  for (int t = 0; t < CH; ++t) {
    size_t tok = (size_t)b * L_ + z * CH + t;
    acc += laB[tok * NH + h];
    Acum[tok * NH + h] = acc;
  }
  cd[((size_t)b * NC + z) * NH + h] = expf(acc);
}

// rope(rms + bias) single element (n in [0,128))
static __device__ __forceinline__ float rope_elem(
    const float* __restrict__ rms_row, const float* __restrict__ bias,
    const float* __restrict__ cs, const float* __restrict__ sn, int n) {
  if (n < NR) {
    float v1 = rms_row[n] + bias[n];
    float v2 = rms_row[n + NR] + bias[n + NR];
    return v1 * cs[n] - v2 * sn[n];
  } else if (n < 2 * NR) {
    int r = n - NR;
    float v1 = rms_row[r] + bias[r];
    float v2 = rms_row[n] + bias[n];
    return v1 * sn[r] + v2 * cs[r];
  }
  return rms_row[n] + bias[n];
}

// -------------------------------------------------------------------------
// Chunk intra kernel: per (b,z,h) workgroup (4 waves) computes
//   CB1/CB2 -> masked M1/M2 -> Yintra, and U (64x128) for the scan.
// X chunks stored transposed (XT[p][s]) so Yintra's B operand and U's A
// operand are k-contiguous 8-byte LDS reads.
// -------------------------------------------------------------------------
struct SharedK4 {
  union {
    float Cn[64][132];
    float M[2][64][66];  // reuse Cn space after CB pass
  } cm;
  float Bn[64][132];
  float Bp[64][132];
  float XcT[64][66];     // XcT[p][s] = x[s][p]
  float XpT[64][66];
  float Ac[64], W1s[64], W2s[64], RW1[64], RW2[64];
};

__global__ __launch_bounds__(128) void chunk_intra_kernel(
    const float* __restrict__ proj, const float* __restrict__ Brms,
    const float* __restrict__ Crms, const float* __restrict__ Bbias,
    const float* __restrict__ Cbias, const float* __restrict__ cosT,
    const float* __restrict__ sinT, const float* __restrict__ Acum,
    const float* __restrict__ w1b, const float* __restrict__ w2b,
    float* __restrict__ Yin, float* __restrict__ U) {
  __shared__ SharedK4 sh;
  const int h = blockIdx.x, z = blockIdx.y, b = blockIdx.z;
  const int tid = threadIdx.x;
  const int wave = tid >> 5, lane = tid & 31;
  const int lm = lane & 15, kg = (lane >> 4) << 1, hi8 = (lane >> 4) << 3;

  if (tid < 64) {
    size_t tok = (size_t)b * L_ + z * CH + tid;
    sh.Ac[tid]  = Acum[tok * NH + h];
    sh.W1s[tid] = w1b[tok * NH + h];
    sh.W2s[tid] = w2b[tok * NH + h];
  }
  // Cn, Bn, Bp (rope-applied), 64x128 each
  for (int i = 0; i < 64; ++i) {
    int idx = tid + i * 128;
    int t = idx >> 7, n = idx & 127;
    size_t tok = (size_t)b * L_ + z * CH + t;
    const float* csr = cosT + (tok * NH + h) * NR;
    const float* snr = sinT + (tok * NH + h) * NR;
    sh.cm.Cn[t][n] = rope_elem(Crms + tok * DS, Cbias + (size_t)h * DS, csr, snr, n);
    sh.Bn[t][n]    = rope_elem(Brms + tok * DS, Bbias + (size_t)h * DS, csr, snr, n);
    if (z == 0 && t == 0) {
      sh.Bp[t][n] = 0.f;
    } else {
      size_t tkp = tok - 1;
      const float* csp = cosT + (tkp * NH + h) * NR;
      const float* snp = sinT + (tkp * NH + h) * NR;
      sh.Bp[t][n] = rope_elem(Brms + tkp * DS, Bbias + (size_t)h * DS, csp, snp, n);
    }
  }
  // Xc, Xp stored transposed: XT[p][t] = x[t][p]; global reads coalesced in p
  for (int i = 0; i < 32; ++i) {
    int idx = tid + i * 128;
    int t = idx >> 6, p = idx & 63;
    size_t tok = (size_t)b * L_ + z * CH + t;
    sh.XcT[p][t] = proj[tok * DIP + h * HD + p];
    sh.XpT[p][t] = (z == 0 && t == 0) ? 0.f : proj[(tok - 1) * DIP + h * HD + p];
  }
  __syncthreads();
  if (tid < 64) {
    float rr = expf(sh.Ac[63] - sh.Ac[tid]);
    sh.RW1[tid] = rr * sh.W1s[tid];
    sh.RW2[tid] = rr * sh.W2s[tid];
  }
  __syncthreads();

  // ---- CB1 = Cn * Bn^T, CB2 = Cn * Bp^T  (64x64, K=128) ----
  v8f cb1[4], cb2[4];
#pragma unroll
  for (int t = 0; t < 4; ++t) { cb1[t] = v8zero(); cb2[t] = v8zero(); }
  for (int kk = 0; kk < DS; kk += 4) {
    const int trow = wave * 16 + lm;
    v2f a = *(const v2f*)(&sh.cm.Cn[trow][kk + kg]);
#pragma unroll
    for (int t = 0; t < 4; ++t) {
      const int sc = t * 16 + lm;
      v2f b1 = *(const v2f*)(&sh.Bn[sc][kk + kg]);
      cb1[t] = wmma4(a, b1, cb1[t]);
      v2f b2 = *(const v2f*)(&sh.Bp[sc][kk + kg]);
      cb2[t] = wmma4(a, b2, cb2[t]);
    }
  }
  __syncthreads();  // everyone done reading Cn; its LDS becomes M1/M2
  // ---- apply decay mask + weights, store M1/M2 ----
#pragma unroll
  for (int t = 0; t < 4; ++t) {
    const int sc = t * 16 + lm;
#pragma unroll
    for (int i = 0; i < 8; ++i) {
      int trow = wave * 16 + i + hi8;
      float d = (trow >= sc) ? expf(sh.Ac[trow] - sh.Ac[sc]) : 0.f;
      sh.cm.M[0][trow][sc] = d * cb1[t][i] * sh.W1s[sc];
      sh.cm.M[1][trow][sc] = d * cb2[t][i] * sh.W2s[sc];
    }
  }
  __syncthreads();

  // ---- Yintra = M1*Xc + M2*Xp  (64x64, K=64) ----
  v8f yi[4];
#pragma unroll
  for (int t = 0; t < 4; ++t) yi[t] = v8zero();
  for (int kk = 0; kk < CH; kk += 4) {
    const int trow = wave * 16 + lm;
    v2f a1 = *(const v2f*)(&sh.cm.M[0][trow][kk + kg]);
    v2f a2 = *(const v2f*)(&sh.cm.M[1][trow][kk + kg]);
#pragma unroll
    for (int t = 0; t < 4; ++t) {
      const int pc = t * 16 + lm;
      v2f b1 = *(const v2f*)(&sh.XcT[pc][kk + kg]);
      yi[t] = wmma4(a1, b1, yi[t]);
      v2f b2 = *(const v2f*)(&sh.XpT[pc][kk + kg]);
      yi[t] = wmma4(a2, b2, yi[t]);
    }
  }
#pragma unroll
  for (int t = 0; t < 4; ++t)
#pragma unroll
    for (int i = 0; i < 8; ++i) {
      int trow = wave * 16 + i + hi8;
      int p = t * 16 + lm;
      size_t tok = (size_t)b * L_ + z * CH + trow;
      Yin[(tok * NH + h) * HD + p] = yi[t][i];
    }

  // ---- U[p,n] = sum_s rw1[s]*Xc[s,p]*Bn[s,n] + rw2[s]*Xp[s,p]*Bp[s,n] ----
  v8f ua[8];
#pragma unroll
  for (int t = 0; t < 8; ++t) ua[t] = v8zero();
  for (int kk = 0; kk < CH; kk += 4) {
    const int prow = wave * 16 + lm;
    v2f a1 = *(const v2f*)(&sh.XcT[prow][kk + kg]);
    v2f a2 = *(const v2f*)(&sh.XpT[prow][kk + kg]);
    a1.x *= sh.RW1[kk + kg];
    a1.y *= sh.RW1[kk + kg + 1];
    a2.x *= sh.RW2[kk + kg];
    a2.y *= sh.RW2[kk + kg + 1];
#pragma unroll
    for (int t = 0; t < 8; ++t) {
      const int nn = t * 16 + lm;
      v2f b1; b1.x = sh.Bn[kk + kg][nn]; b1.y = sh.Bn[kk + kg + 1][nn];
      ua[t] = wmma4(a1, b1, ua[t]);
      v2f b2; b2.x = sh.Bp[kk + kg][nn]; b2.y = sh.Bp[kk + kg + 1][nn];
      ua[t] = wmma4(a2, b2, ua[t]);
    }
  }
  size_t ub = (((size_t)b * NC + z) * NH + h) * (size_t)(HD * DS);
#pragma unroll
  for (int t = 0; t < 8; ++t)
#pragma unroll
    for (int i = 0; i < 8; ++i) {
      int prow = wave * 16 + i + hi8;
      int n = t * 16 + lm;
      U[ub + (size_t)prow * DS + n] = ua[t][i];
    }
}

// -------------------------------------------------------------------------
// Sequential chunk scan (in place): states[z] = carry; carry = cd*carry + U[z]
// U buffer is overwritten with states.  One block per (b,h).
// -------------------------------------------------------------------------
__global__ __launch_bounds__(256) void scan_kernel(float* __restrict__ U,
                                                   const float* __restrict__ cd) {
  const int bh = blockIdx.x;
  const int b = bh / NH, h = bh % NH;
  const int tid = threadIdx.x;
  float hr[32];
#pragma unroll
  for (int j = 0; j < 32; ++j) hr[j] = 0.f;
  for (int z = 0; z < NC; ++z) {
    float c = cd[((size_t)b * NC + z) * NH + h];
    size_t base = (((size_t)b * NC + z) * NH + h) * (size_t)(HD * DS);
#pragma unroll
    for (int j = 0; j < 32; ++j) {
      size_t g = base + tid + j * 256;
      float uv = U[g];
      U[g] = hr[j];
      hr[j] = c * hr[j] + uv;
    }
  }
}

// -------------------------------------------------------------------------
// Ystate = (Cn * exp(Acum)) @ states^T ; y = Yintra + Ystate + x*D
// -------------------------------------------------------------------------
struct SharedK6 {
  float Cn[64][132];
  float St[64][132];  // St[p][n]
  float eA[64];
};

__global__ __launch_bounds__(128) void ystate_kernel(
    const float* __restrict__ proj, const float* __restrict__ Crms,
    const float* __restrict__ Cbias, const float* __restrict__ cosT,
    const float* __restrict__ sinT, const float* __restrict__ Acum,
    const float* __restrict__ states, const float* __restrict__ Yin,
    const float* __restrict__ Dv, float* __restrict__ y) {
  __shared__ SharedK6 sh;
  const int h = blockIdx.x, z = blockIdx.y, b = blockIdx.z;
  const int tid = threadIdx.x;
  const int wave = tid >> 5, lane = tid & 31;
  const int lm = lane & 15, kg = (lane >> 4) << 1, hi8 = (lane >> 4) << 3;

  if (tid < 64) {
    size_t tok = (size_t)b * L_ + z * CH + tid;
    sh.eA[tid] = expf(Acum[tok * NH + h]);
  }
  size_t sb = (((size_t)b * NC + z) * NH + h) * (size_t)(HD * DS);
  for (int i = 0; i < 64; ++i) {
    int idx = tid + i * 128;
    int t = idx >> 7, n = idx & 127;
    size_t tok = (size_t)b * L_ + z * CH + t;
    sh.Cn[t][n] = rope_elem(Crms + tok * DS, Cbias + (size_t)h * DS,
                            cosT + (tok * NH + h) * NR,
                            sinT + (tok * NH + h) * NR, n);
    sh.St[t][n] = states[sb + (size_t)t * DS + n];  // p=t
  }
  __syncthreads();

  v8f acc[4];
#pragma unroll
  for (int t = 0; t < 4; ++t) acc[t] = v8zero();
  for (int kk = 0; kk < DS; kk += 4) {
    const int trow = wave * 16 + lm;
    float e = sh.eA[trow];
    v2f a = *(const v2f*)(&sh.Cn[trow][kk + kg]);
    a.x *= e;
    a.y *= e;
#pragma unroll
    for (int t = 0; t < 4; ++t) {
      const int pc = t * 16 + lm;
      v2f bf = *(const v2f*)(&sh.St[pc][kk + kg]);
      acc[t] = wmma4(a, bf, acc[t]);
    }
  }
  float dv = Dv[h];
#pragma unroll
  for (int t = 0; t < 4; ++t)
#pragma unroll
    for (int i = 0; i < 8; ++i) {
      int trow = wave * 16 + i + hi8;
      int p = t * 16 + lm;
      size_t tok = (size_t)b * L_ + z * CH + trow;
      float xv = proj[tok * DIP + h * HD + p];
      y[tok * DI + h * HD + p] =
          Yin[(tok * NH + h) * HD + p] + acc[t][i] + xv * dv;
    }
}

// -------------------------------------------------------------------------
extern "C" void kernel_launch(void* const* d_in, const int* in_sizes, int n_in,
                              void* d_out, int out_size, void* d_ws, size_t ws_size,
                              hipStream_t stream) {
  (void)in_sizes; (void)n_in; (void)out_size; (void)ws_size;
  const float* u       = (const float*)d_in[0];
  const float* W_in    = (const float*)d_in[1];
  const float* dt_bias = (const float*)d_in[2];
  const float* B_bias  = (const float*)d_in[3];
  const float* C_bias  = (const float*)d_in[4];
  const float* Bnw     = (const float*)d_in[5];
  const float* Cnw     = (const float*)d_in[6];
  const float* Dv      = (const float*)d_in[7];
  const float* W_out   = (const float*)d_in[8];
  float* out = (float*)d_out;

  float* ws = (float*)d_ws;
  size_t off = 0;
  float* proj = ws + off; off += (size_t)B_ * L_ * DIP;
  float* Brms = ws + off; off += (size_t)B_ * L_ * DS;
  float* Crms = ws + off; off += (size_t)B_ * L_ * DS;
  float* laB  = ws + off; off += (size_t)B_ * L_ * NH;
  float* DTb  = ws + off; off += (size_t)B_ * L_ * NH;
  float* w1b  = ws + off; off += (size_t)B_ * L_ * NH;
  float* w2b  = ws + off; off += (size_t)B_ * L_ * NH;
  float* Acum = ws + off; off += (size_t)B_ * L_ * NH;
  float* cosT = ws + off; off += (size_t)B_ * L_ * NH * NR;
  float* sinT = ws + off; off += (size_t)B_ * L_ * NH * NR;
  float* U    = ws + off; off += (size_t)B_ * NC * NH * HD * DS;
  float* cd   = ws + off; off += (size_t)B_ * NC * NH;
  float* Yin  = ws + off; off += (size_t)B_ * L_ * NH * HD;
  float* ybuf = ws + off; off += (size_t)B_ * L_ * DI;

  // 1) in-projection GEMM: proj = u * W_in^T
  gemm_nt_kernel<<<dim3(DIP / 64, (B_ * L_) / 128), 128, 0, stream>>>(
      u, W_in, proj, B_ * L_, DIP, DM);
  // 2) per-token prep
  prep_token_kernel<<<B_ * L_, 128, 0, stream>>>(proj, dt_bias, Bnw, Cnw, Brms,
                                                 Crms, laB, DTb, w1b, w2b);
  // 3) theta cumsum -> cos/sin
  theta_kernel<<<(B_ * NH * NR + 255) / 256, 256, 0, stream>>>(proj, DTb, cosT,
                                                               sinT);
  // 4) per-chunk Acum + cd
  acum_kernel<<<(B_ * NC * NH + 255) / 256, 256, 0, stream>>>(laB, Acum, cd);
  // 5) chunk intra: Yintra + U
  chunk_intra_kernel<<<dim3(NH, NC, B_), 128, 0, stream>>>(
      proj, Brms, Crms, B_bias, C_bias, cosT, sinT, Acum, w1b, w2b, Yin, U);
  // 6) sequential chunk scan (U becomes states in place)
  scan_kernel<<<B_ * NH, 256, 0, stream>>>(U, cd);
  // 7) Ystate + combine -> y
  ystate_kernel<<<dim3(NH, NC, B_), 128, 0, stream>>>(
      proj, Crms, C_bias, cosT, sinT, Acum, U, Yin, Dv, ybuf);
  // 8) out-projection GEMM: out = y * W_out^T
  gemm_nt_kernel<<<dim3(DM / 64, (B_ * L_) / 128), 128, 0, stream>>>(
      ybuf, W_out, out, B_ * L_, DM, DI);
}